// PTKGauss_65369402245390
// MI455X (gfx1250) — compile-verified
//
#include <hip/hip_runtime.h>

typedef float v2f __attribute__((ext_vector_type(2)));
typedef float v8f __attribute__((ext_vector_type(8)));

#define TILE 128            // output tile per block (128x128)
#define DK 64               // feature dimension
#define SA_STRIDE 68        // LDS row stride (floats): 272B, 16B-aligned, bank-spread
#define OUT_STRIDE 132      // LDS out-stage row stride (floats): 528B, 16B-aligned
#define SA_ELEMS (TILE * SA_STRIDE)          // 8704 floats per input tile
#define NORM_OFF (2 * SA_ELEMS)              // 17408 floats
#define SMEM_FLOATS (NORM_OFF + 2 * TILE)    // + 256 norm floats = 17664
#define SMEM_BYTES (SMEM_FLOATS * 4)         // 70656 B < 320 KB WGP LDS

__global__ __launch_bounds__(256)
void gauss_gram_wmma_kernel(const float* __restrict__ X,
                            float* __restrict__ K,
                            int n) {
    extern __shared__ float smem[];
    float* sA   = smem;                  // [TILE][SA_STRIDE]
    float* sB   = smem + SA_ELEMS;       // [TILE][SA_STRIDE]
    float* sNA  = smem + NORM_OFF;       // [TILE]
    float* sNB  = sNA + TILE;            // [TILE]
    float* sOut = smem;                  // overlay of sA/sB after compute: [TILE][OUT_STRIDE]

    const int tid   = threadIdx.x;
    const int lane  = tid & 31;
    const int wave  = tid >> 5;          // 0..7
    const int hi    = lane >> 4;         // 0: K pair {k0,k0+1}; 1: {k0+2,k0+3}
    const int r     = lane & 15;         // row (A) / col (B) within 16-tile
    const int waveM = wave >> 2;         // 0..1  -> 64-row strip
    const int waveN = wave & 3;          // 0..3  -> 32-col strip
    const int mBase = waveM * 64;
    const int nBase = waveN * 32;

    const int rowBlock = blockIdx.y * TILE;
    const int colBlock = blockIdx.x * TILE;

    // ---- Stage A (rows) and B (cols-as-rows of X) tiles into LDS, float4 ----
    for (int i = tid; i < TILE * (DK / 4); i += 256) {
        const int row = i >> 4;
        const int c4  = (i & 15) * 4;
        float4 a = *reinterpret_cast<const float4*>(X + (size_t)(rowBlock + row) * DK + c4);
        float4 b = *reinterpret_cast<const float4*>(X + (size_t)(colBlock + row) * DK + c4);
        *reinterpret_cast<float4*>(sA + row * SA_STRIDE + c4) = a;
        *reinterpret_cast<float4*>(sB + row * SA_STRIDE + c4) = b;
    }
    __syncthreads();

    // ---- Row norms (one row per thread; 128 A rows + 128 B rows) ----
    {
        const float* src = (tid < TILE) ? (sA + tid * SA_STRIDE)
                                        : (sB + (tid - TILE) * SA_STRIDE);
        float s = 0.0f;
#pragma unroll
        for (int k = 0; k < DK; ++k) s = fmaf(src[k], src[k], s);
        ((tid < TILE) ? sNA : sNB)[tid & (TILE - 1)] = s;
    }

    // ---- WMMA main loop: 64x32 register tile per wave, f32 16x16x4 ----
    v8f acc[4][2] = {};
    for (int k0 = 0; k0 < DK; k0 += 4) {
        const int kk = k0 + 2 * hi;
        v2f af[4], bf[2];
#pragma unroll
        for (int tm = 0; tm < 4; ++tm)
            af[tm] = *reinterpret_cast<const v2f*>(sA + (mBase + tm * 16 + r) * SA_STRIDE + kk);
#pragma unroll
        for (int tn = 0; tn < 2; ++tn)
            bf[tn] = *reinterpret_cast<const v2f*>(sB + (nBase + tn * 16 + r) * SA_STRIDE + kk);
#pragma unroll
        for (int tm = 0; tm < 4; ++tm)
#pragma unroll
            for (int tn = 0; tn < 2; ++tn)
                acc[tm][tn] = __builtin_amdgcn_wmma_f32_16x16x4_f32(
                    /*neg_a=*/false, af[tm],
                    /*neg_b=*/false, bf[tn],
                    /*c_mod=*/(short)0, acc[tm][tn],
                    /*reuse_a=*/false, /*reuse_b=*/false);
    }

    // All LDS tile reads done (and norm writes globally visible) after this barrier;
    // sOut overlays sA/sB below.
    __syncthreads();

    // ---- Epilogue: d2 = max(|xi|^2 + |xj|^2 - 2*dot, 0); K = exp(-d2/2); stage to LDS ----
#pragma unroll
    for (int tn = 0; tn < 2; ++tn) {
        const int col = nBase + tn * 16 + r;
        const float nj = sNB[col];
#pragma unroll
        for (int tm = 0; tm < 4; ++tm) {
            const int rowBase = mBase + tm * 16 + 8 * hi;
#pragma unroll
            for (int v = 0; v < 8; ++v) {
                const int row = rowBase + v;
                const float ni = sNA[row];
                float d2 = fmaf(-2.0f, acc[tm][tn][v], ni + nj);
                d2 = fmaxf(d2, 0.0f);
                sOut[row * OUT_STRIDE + col] = __expf(-0.5f * d2);
            }
        }
    }
    __syncthreads();

    // ---- Coalesced store: one 512B row per wave-iteration (32 lanes x float4) ----
    for (int it = 0; it < 16; ++it) {
        const int lrow = wave * 16 + it;
        const int grow = rowBlock + lrow;
        float4 v = *reinterpret_cast<const float4*>(sOut + lrow * OUT_STRIDE + lane * 4);
        *reinterpret_cast<float4*>(K + (size_t)grow * n + colBlock + lane * 4) = v;
    }
}

extern "C" void kernel_launch(void* const* d_in, const int* in_sizes, int n_in,
                              void* d_out, int out_size, void* d_ws, size_t ws_size,
                              hipStream_t stream) {
    (void)n_in; (void)d_ws; (void)ws_size; (void)out_size;
    const float* X = (const float*)d_in[0];
    float* K = (float*)d_out;
    const int n = in_sizes[0] / DK;          // 16384
    dim3 grid(n / TILE, n / TILE);           // 128 x 128 blocks
    dim3 block(256);
    gauss_gram_wmma_kernel<<<grid, block, SMEM_BYTES, stream>>>(X, K, n);
}